// GAT_21002390078215
// MI455X (gfx1250) — compile-verified
//
#include <hip/hip_runtime.h>
#include <math.h>

typedef __attribute__((ext_vector_type(2))) float v2f;
typedef __attribute__((ext_vector_type(8))) float v8f;

#define N_NODES 32768
#define NE_NODES 16
#define NM_NODES 32752
#define N_EDGES 524288

__device__ __forceinline__ float leakyf(float x, float s) { return x > 0.f ? x : s * x; }

__device__ __forceinline__ void atomicMaxF(float* a, float v) {
    // Monotonic bit-pattern trick: works for all finite floats & +/-inf
    if (v >= 0.f) atomicMax((int*)a, __float_as_int(v));
    else          atomicMin((unsigned int*)a, __float_as_uint(v));
}

// ---------------- CNN front end ----------------
// conv1: [16,2,1600,1600] -> [16,2,400,400], k=4 s=4, BN + LeakyReLU(0.1)
// One thread computes BOTH output channels so each input element is read once.
__global__ void conv1_k(const float* __restrict__ in, const float* __restrict__ w,
                        const float* __restrict__ g, const float* __restrict__ b,
                        float* __restrict__ out) {
    int idx = blockIdx.x * blockDim.x + threadIdx.x;
    const int HW = 400 * 400;
    if (idx >= NE_NODES * HW) return;
    int ox = idx % 400, oy = (idx / 400) % 400, n = idx / HW;
    float a0 = 0.f, a1 = 0.f;
    const float* base = in + (size_t)n * 2 * 1600 * 1600;
#pragma unroll
    for (int c = 0; c < 2; c++) {
        const float* p = base + (size_t)c * 1600 * 1600 + (size_t)(oy * 4) * 1600 + ox * 4;
#pragma unroll
        for (int r = 0; r < 4; r++) {
            float4 v = *(const float4*)(p + (size_t)r * 1600);   // global_load_b128
            const float* w0 = w + ((0 * 2 + c) * 4 + r) * 4;
            const float* w1 = w + ((1 * 2 + c) * 4 + r) * 4;
            a0 = fmaf(v.x, w0[0], fmaf(v.y, w0[1], fmaf(v.z, w0[2], fmaf(v.w, w0[3], a0))));
            a1 = fmaf(v.x, w1[0], fmaf(v.y, w1[1], fmaf(v.z, w1[2], fmaf(v.w, w1[3], a1))));
        }
    }
    float invs = rsqrtf(1.0f + 1e-5f);
    out[((size_t)(n * 2 + 0) * 400 + oy) * 400 + ox] = leakyf(a0 * (g[0] * invs) + b[0], 0.1f);
    out[((size_t)(n * 2 + 1) * 400 + oy) * 400 + ox] = leakyf(a1 * (g[1] * invs) + b[1], 0.1f);
}

// conv2: [16,2,400,400] -> [16,100,100]
__global__ void conv2_k(const float* __restrict__ in, const float* __restrict__ w,
                        const float* __restrict__ g, const float* __restrict__ b,
                        float* __restrict__ out) {
    int idx = blockIdx.x * blockDim.x + threadIdx.x;
    const int HW = 100 * 100;
    if (idx >= NE_NODES * HW) return;
    int ox = idx % 100, oy = (idx / 100) % 100, n = idx / HW;
    float a = 0.f;
#pragma unroll
    for (int c = 0; c < 2; c++) {
        const float* p = in + ((size_t)(n * 2 + c) * 400 + oy * 4) * 400 + ox * 4;
#pragma unroll
        for (int r = 0; r < 4; r++) {
            float4 v = *(const float4*)(p + (size_t)r * 400);
            const float* wr = w + (c * 4 + r) * 4;
            a = fmaf(v.x, wr[0], fmaf(v.y, wr[1], fmaf(v.z, wr[2], fmaf(v.w, wr[3], a))));
        }
    }
    float invs = rsqrtf(1.0f + 1e-5f);
    out[(size_t)n * HW + oy * 100 + ox] = leakyf(a * (g[0] * invs) + b[0], 0.1f);
}

// conv3: [16,100,100] -> [16,625]  (flattened 25x25)
__global__ void conv3_k(const float* __restrict__ in, const float* __restrict__ w,
                        const float* __restrict__ g, const float* __restrict__ b,
                        float* __restrict__ out) {
    int idx = blockIdx.x * blockDim.x + threadIdx.x;
    const int HW = 25 * 25;
    if (idx >= NE_NODES * HW) return;
    int ox = idx % 25, oy = (idx / 25) % 25, n = idx / HW;
    float a = 0.f;
    const float* p = in + (size_t)n * 10000 + (size_t)(oy * 4) * 100 + ox * 4;
#pragma unroll
    for (int r = 0; r < 4; r++) {
        float4 v = *(const float4*)(p + (size_t)r * 100);
        const float* wr = w + r * 4;
        a = fmaf(v.x, wr[0], fmaf(v.y, wr[1], fmaf(v.z, wr[2], fmaf(v.w, wr[3], a))));
    }
    float invs = rsqrtf(1.0f + 1e-5f);
    out[(size_t)n * HW + oy * 25 + ox] = leakyf(a * (g[0] * invs) + b[0], 0.1f);
}

// Enzyme head: x3[16,625]@lin625^T, e_single[16,1600]@single^T -> concat[128] -> fc0[:, :128]
__global__ void enzyme_head_k(const float* __restrict__ x3, const float* __restrict__ esingle,
                              const float* __restrict__ lw, const float* __restrict__ lb,
                              const float* __restrict__ sw, const float* __restrict__ sb,
                              const float* __restrict__ f0w, const float* __restrict__ f0b,
                              float* __restrict__ h) {
    __shared__ float ef[128];
    int n = blockIdx.x, t = threadIdx.x;
    if (t < 64) {
        float acc = lb[t];
        const float* xr = x3 + n * 625;
        const float* wr = lw + t * 625;
        for (int k = 0; k < 625; k++) acc = fmaf(xr[k], wr[k], acc);
        ef[t] = acc;
    } else {
        int j = t - 64;
        float acc = sb[j];
        const float* xr = esingle + n * 1600;
        const float* wr = sw + j * 1600;
        for (int k = 0; k < 1600; k++) acc = fmaf(xr[k], wr[k], acc);
        ef[t] = acc;
    }
    __syncthreads();
    if (t < 64) {
        float acc = f0b[t];
        const float* wr = f0w + t * 257;   // only first 128 cols are non-zero-multiplied
        for (int k = 0; k < 128; k++) acc = fmaf(ef[k], wr[k], acc);
        h[n * 64 + t] = acc;
    }
}

// Metabolite projection: m_feat[32752,129] @ fc1_w[:,128:257]^T + fc1_b -> h rows 16..32767
__global__ void metab_k(const float* __restrict__ mfeat, const float* __restrict__ f1w,
                        const float* __restrict__ f1b, float* __restrict__ h) {
    __shared__ float row[129];
    int i = blockIdx.x, t = threadIdx.x;
    for (int k = t; k < 129; k += 64) row[k] = mfeat[(size_t)i * 129 + k];
    __syncthreads();
    float acc = f1b[t];
    const float* wr = f1w + t * 257 + 128;
    for (int k = 0; k < 129; k++) acc = fmaf(row[k], wr[k], acc);
    h[(size_t)(NE_NODES + i) * 64 + t] = acc;
}

// ---------------- fp32 WMMA GEMM: C[M,N] = A[M,K] @ B[K,N] ----------------
// One wave computes a 16 x (16*NACC) tile of C with V_WMMA_F32_16X16X4_F32.
// A frag (16x4): lane L -> row L%16, K-pair 2*(L/16); B frag (4x16) mirrored;
// C/D: row = vgpr + 8*(L/16), col = L%16.
template <int NACC>
__global__ __launch_bounds__(256) void wmma_gemm_k(const float* __restrict__ A,
                                                   const float* __restrict__ B,
                                                   float* __restrict__ C,
                                                   int M, int N, int K) {
    int gtid = blockIdx.x * blockDim.x + threadIdx.x;
    int wave = gtid >> 5;
    int lane = threadIdx.x & 31;
    int tilesN = N / (16 * NACC);
    int tm = wave / tilesN;
    int tnb = (wave - tm * tilesN) * (16 * NACC);
    if (tm * 16 >= M) return;                      // uniform per wave
    int mrow = tm * 16 + (lane & 15);
    int kh = (lane >> 4) * 2;                      // 0 or 2
    const float* arow = A + (size_t)mrow * K;
    int nc = tnb + (lane & 15);
    v8f acc[NACC];
#pragma unroll
    for (int j = 0; j < NACC; j++)
#pragma unroll
        for (int v = 0; v < 8; v++) acc[j][v] = 0.f;
    for (int k0 = 0; k0 < K; k0 += 4) {
        v2f a;
        a.x = arow[k0 + kh];
        a.y = arow[k0 + kh + 1];
#pragma unroll
        for (int j = 0; j < NACC; j++) {
            v2f bb;
            bb.x = B[(size_t)(k0 + kh) * N + nc + j * 16];
            bb.y = B[(size_t)(k0 + kh + 1) * N + nc + j * 16];
            acc[j] = __builtin_amdgcn_wmma_f32_16x16x4_f32(
                false, a, false, bb, (short)0, acc[j], false, false);
        }
    }
    int mb = tm * 16 + 8 * (lane >> 4);
#pragma unroll
    for (int j = 0; j < NACC; j++)
#pragma unroll
        for (int v = 0; v < 8; v++)
            C[(size_t)(mb + v) * N + tnb + j * 16 + (lane & 15)] = acc[j][v];
}

// ---------------- GAT edge phase ----------------
// el[n,h] = sum_d f[n,h,d]*al[h,d]; er likewise. One wave per (node, head).
__global__ void attn_k(const float* __restrict__ f, const float* __restrict__ al,
                       const float* __restrict__ ar, float* __restrict__ el,
                       float* __restrict__ er, int H, int D) {
    int wave = (blockIdx.x * blockDim.x + threadIdx.x) >> 5;
    int lane = threadIdx.x & 31;
    int node = wave / H, h = wave - node * H;
    if (node >= N_NODES) return;
    float sl = 0.f, sr = 0.f;
    const float* fr = f + (size_t)node * H * D + h * D;
    for (int d = lane; d < D; d += 32) {
        float fv = fr[d];
        sl = fmaf(fv, al[h * D + d], sl);
        sr = fmaf(fv, ar[h * D + d], sr);
    }
    for (int off = 16; off > 0; off >>= 1) {
        sl += __shfl_down(sl, off);
        sr += __shfl_down(sr, off);
    }
    if (lane == 0) { el[node * H + h] = sl; er[node * H + h] = sr; }
}

__global__ void edge_max_k(const float* __restrict__ el, const float* __restrict__ er,
                           const int* __restrict__ src, const int* __restrict__ dst,
                           float* __restrict__ emax, int H) {
    int i = blockIdx.x * blockDim.x + threadIdx.x;
    if (i >= N_EDGES * H) return;
    int e = i / H, h = i - e * H;
    int s = src[e], d = dst[e];
    float v = el[s * H + h] + er[d * H + h];
    v = leakyf(v, 0.2f);
    atomicMaxF(&emax[d * H + h], v);
}

__global__ void edge_exp_k(const float* __restrict__ el, const float* __restrict__ er,
                           const int* __restrict__ src, const int* __restrict__ dst,
                           const float* __restrict__ emax, float* __restrict__ ex,
                           float* __restrict__ den, int H) {
    int i = blockIdx.x * blockDim.x + threadIdx.x;
    if (i >= N_EDGES * H) return;
    int e = i / H, h = i - e * H;
    int s = src[e], d = dst[e];
    float v = el[s * H + h] + er[d * H + h];
    v = leakyf(v, 0.2f);
    float x = __expf(v - emax[d * H + h]);
    ex[i] = x;
    atomicAdd(&den[d * H + h], x);
}

// One wave per edge: gather f[src,:] (coalesced), scale by alpha, scatter-add to out[dst,:]
__global__ void edge_agg_k(const float* __restrict__ f, const int* __restrict__ src,
                           const int* __restrict__ dst, const float* __restrict__ ex,
                           const float* __restrict__ den, float* __restrict__ out,
                           int H, int D) {
    int wave = (blockIdx.x * blockDim.x + threadIdx.x) >> 5;
    int lane = threadIdx.x & 31;
    if (wave >= N_EDGES) return;
    int s = src[wave], d = dst[wave];
    int HD = H * D;
    const float* fr = f + (size_t)s * HD;
    float* orow = out + (size_t)d * HD;
    for (int idx = lane; idx < HD; idx += 32) {
        int h = idx / D;
        float alpha = ex[(size_t)wave * H + h] / fmaxf(den[d * H + h], 1e-9f);
        atomicAdd(&orow[idx], fr[idx] * alpha);   // global_atomic_add_f32 (no return)
    }
}

__global__ void elu_k(float* __restrict__ p, int n) {
    int i = blockIdx.x * blockDim.x + threadIdx.x;
    if (i < n) { float x = p[i]; p[i] = x > 0.f ? x : (__expf(x) - 1.f); }
}

__global__ void fill_k(float* __restrict__ p, float v, int n) {
    int i = blockIdx.x * blockDim.x + threadIdx.x;
    if (i < n) p[i] = v;
}

extern "C" void kernel_launch(void* const* d_in, const int* in_sizes, int n_in,
                              void* d_out, int out_size, void* d_ws, size_t ws_size,
                              hipStream_t stream) {
    const float* e_logits = (const float*)d_in[0];
    const float* e_single = (const float*)d_in[1];
    const float* m_feat   = (const float*)d_in[2];
    const int* src = (const int*)d_in[3];
    const int* dst = (const int*)d_in[4];
    const float* w1 = (const float*)d_in[5];
    const float* w2 = (const float*)d_in[6];
    const float* w3 = (const float*)d_in[7];
    const float* g1 = (const float*)d_in[8];  const float* b1 = (const float*)d_in[9];
    const float* g2 = (const float*)d_in[10]; const float* b2 = (const float*)d_in[11];
    const float* g3 = (const float*)d_in[12]; const float* b3 = (const float*)d_in[13];
    const float* lw = (const float*)d_in[14]; const float* lb = (const float*)d_in[15];
    const float* sw = (const float*)d_in[16]; const float* sb = (const float*)d_in[17];
    const float* f0w = (const float*)d_in[18]; const float* f0b = (const float*)d_in[19];
    const float* f1w = (const float*)d_in[20]; const float* f1b = (const float*)d_in[21];
    const float* W0 = (const float*)d_in[22];
    const float* al0 = (const float*)d_in[23]; const float* ar0 = (const float*)d_in[24];
    const float* W1 = (const float*)d_in[25];
    const float* al1 = (const float*)d_in[26]; const float* ar1 = (const float*)d_in[27];
    const float* W2 = (const float*)d_in[28];
    const float* al2 = (const float*)d_in[29]; const float* ar2 = (const float*)d_in[30];
    float* outp = (float*)d_out;

    float* ws = (float*)d_ws;
    size_t off = 0;
    auto alloc = [&](size_t n) { float* p = ws + off; off += n; return p; };
    float* fbuf = alloc((size_t)N_NODES * 256);   // also conv1 output x1 (5.12M < 8.39M floats)
    float* gbuf = alloc((size_t)N_NODES * 256);   // aggregation / next-layer input
    float* x2   = alloc(160000);
    float* x3   = alloc(10000);
    float* h0   = alloc((size_t)N_NODES * 64);
    float* el   = alloc((size_t)N_NODES * 4);
    float* er   = alloc((size_t)N_NODES * 4);
    float* emax = alloc((size_t)N_NODES * 4);
    float* den  = alloc((size_t)N_NODES * 4);
    float* exb  = alloc((size_t)N_EDGES * 4);

    // ---- CNN + projections ----
    conv1_k<<<(NE_NODES * 400 * 400 + 255) / 256, 256, 0, stream>>>(e_logits, w1, g1, b1, fbuf);
    conv2_k<<<(NE_NODES * 100 * 100 + 255) / 256, 256, 0, stream>>>(fbuf, w2, g2, b2, x2);
    conv3_k<<<(NE_NODES * 25 * 25 + 255) / 256, 256, 0, stream>>>(x2, w3, g3, b3, x3);
    enzyme_head_k<<<NE_NODES, 128, 0, stream>>>(x3, e_single, lw, lb, sw, sb, f0w, f0b, h0);
    metab_k<<<NM_NODES, 64, 0, stream>>>(m_feat, f1w, f1b, h0);

    // ---- GAT layer 0: h0[N,64] @ W0[64,256] ----
    {
        const int H = 4, D = 64, Ncols = 256, K = 64;
        int waves = (N_NODES / 16) * (Ncols / 64);
        wmma_gemm_k<4><<<waves * 32 / 256, 256, 0, stream>>>(h0, W0, fbuf, N_NODES, Ncols, K);
        attn_k<<<N_NODES * H * 32 / 256, 256, 0, stream>>>(fbuf, al0, ar0, el, er, H, D);
        fill_k<<<(N_NODES * H + 255) / 256, 256, 0, stream>>>(emax, -INFINITY, N_NODES * H);
        hipMemsetAsync(den, 0, (size_t)N_NODES * H * sizeof(float), stream);
        edge_max_k<<<(N_EDGES * H + 255) / 256, 256, 0, stream>>>(el, er, src, dst, emax, H);
        edge_exp_k<<<(N_EDGES * H + 255) / 256, 256, 0, stream>>>(el, er, src, dst, emax, exb, den, H);
        hipMemsetAsync(gbuf, 0, (size_t)N_NODES * 256 * sizeof(float), stream);
        edge_agg_k<<<N_EDGES * 32 / 256, 256, 0, stream>>>(fbuf, src, dst, exb, den, gbuf, H, D);
        elu_k<<<(N_NODES * 256 + 255) / 256, 256, 0, stream>>>(gbuf, N_NODES * 256);
    }
    // ---- GAT layer 1: gbuf[N,256] @ W1[256,256] ----
    {
        const int H = 4, D = 64, Ncols = 256, K = 256;
        int waves = (N_NODES / 16) * (Ncols / 64);
        wmma_gemm_k<4><<<waves * 32 / 256, 256, 0, stream>>>(gbuf, W1, fbuf, N_NODES, Ncols, K);
        attn_k<<<N_NODES * H * 32 / 256, 256, 0, stream>>>(fbuf, al1, ar1, el, er, H, D);
        fill_k<<<(N_NODES * H + 255) / 256, 256, 0, stream>>>(emax, -INFINITY, N_NODES * H);
        hipMemsetAsync(den, 0, (size_t)N_NODES * H * sizeof(float), stream);
        edge_max_k<<<(N_EDGES * H + 255) / 256, 256, 0, stream>>>(el, er, src, dst, emax, H);
        edge_exp_k<<<(N_EDGES * H + 255) / 256, 256, 0, stream>>>(el, er, src, dst, emax, exb, den, H);
        hipMemsetAsync(gbuf, 0, (size_t)N_NODES * 256 * sizeof(float), stream);
        edge_agg_k<<<N_EDGES * 32 / 256, 256, 0, stream>>>(fbuf, src, dst, exb, den, gbuf, H, D);
        elu_k<<<(N_NODES * 256 + 255) / 256, 256, 0, stream>>>(gbuf, N_NODES * 256);
    }
    // ---- GAT layer 2: gbuf[N,256] @ W2[256,16]; aggregate straight into d_out ----
    {
        const int H = 1, D = 16, Ncols = 16, K = 256;
        int waves = (N_NODES / 16);
        wmma_gemm_k<1><<<waves * 32 / 256, 256, 0, stream>>>(gbuf, W2, fbuf, N_NODES, Ncols, K);
        attn_k<<<N_NODES * H * 32 / 256, 256, 0, stream>>>(fbuf, al2, ar2, el, er, H, D);
        fill_k<<<(N_NODES * H + 255) / 256, 256, 0, stream>>>(emax, -INFINITY, N_NODES * H);
        hipMemsetAsync(den, 0, (size_t)N_NODES * H * sizeof(float), stream);
        edge_max_k<<<(N_EDGES * H + 255) / 256, 256, 0, stream>>>(el, er, src, dst, emax, H);
        edge_exp_k<<<(N_EDGES * H + 255) / 256, 256, 0, stream>>>(el, er, src, dst, emax, exb, den, H);
        hipMemsetAsync(outp, 0, (size_t)N_NODES * 16 * sizeof(float), stream);
        edge_agg_k<<<N_EDGES * 32 / 256, 256, 0, stream>>>(fbuf, src, dst, exb, den, outp, H, D);
    }
}